// MultisenseLearner_3367254360620
// MI455X (gfx1250) — compile-verified
//
#include <hip/hip_runtime.h>
#include <hip/hip_bf16.h>

// MI455X multisense-learner kernel.
// Memory-bound gather (~0.96 GB @ 23.3 TB/s => ~41us floor, less with L2 reuse).
// 4 batch pairs per wave32; gram matrices via V_WMMA_F32_16X16X4_F32 on the
// block diagonal of a 16x16 accumulator; logsumexp via 4-lane shfl reduction.

typedef __attribute__((ext_vector_type(2))) float v2f;
typedef __attribute__((ext_vector_type(4))) float v4f;
typedef __attribute__((ext_vector_type(8))) float v8f;

#define D_DIM 300
#define S_DIM 4
#define ROW   (D_DIM * S_DIM)   // 1200 floats per vocab row (4.8 KB)

__global__ __launch_bounds__(256)
void multisense_wmma_kernel(const float* __restrict__ V,
                            const float* __restrict__ W,
                            const float* __restrict__ vb,
                            const float* __restrict__ wb,
                            const int*   __restrict__ IJ,
                            float*       __restrict__ out,
                            int B)
{
    const int lane  = threadIdx.x & 31;
    const int wave  = threadIdx.x >> 5;
    const int g     = blockIdx.x * 32 + wave * 4;   // first pair of this wave's group of 4

    const int sub   = (lane & 15) >> 2;  // pair within group (0..3): A's M/4 and B's N/4
    const int s     = lane & 3;          // sense lane: A's M%4 == B's N%4
    const int khalf = lane >> 4;         // 0 -> K={0,1}, 1 -> K={2,3} (ISA f32 A/B layout)

    int pair = g + sub;
    if (pair >= B) pair = B - 1;         // clamp tail: EXEC must remain all-ones for WMMA

    const int iIdx = IJ[2 * pair + 0];   // row of V (senses matrix for columns / t)
    const int jIdx = IJ[2 * pair + 1];   // row of W (senses matrix for rows / s)

    const float* __restrict__ Vrow = V + (size_t)iIdx * ROW;
    const float* __restrict__ Wrow = W + (size_t)jIdx * ROW;

    // Kick the line fills for both rows early (global_prefetch_b8).
    __builtin_prefetch(Wrow + lane * 38, 0, 0);
    __builtin_prefetch(Vrow + lane * 38, 0, 0);

    v8f acc = {};   // 16x16 f32 accumulator: block diagonal = our four 4x4 grams

#if __has_builtin(__builtin_amdgcn_wmma_f32_16x16x4_f32)
    #pragma unroll 5
    for (int d = 0; d < D_DIM; d += 4) {
        const int off = (d + 2 * khalf) * S_DIM + s;
        v2f a, b;
        a.x = Wrow[off];            // A[M][K=2*khalf    ]
        a.y = Wrow[off + S_DIM];    // A[M][K=2*khalf + 1]
        b.x = Vrow[off];            // B[K=2*khalf    ][N]
        b.y = Vrow[off + S_DIM];    // B[K=2*khalf + 1][N]
        acc = __builtin_amdgcn_wmma_f32_16x16x4_f32(
            /*neg_a=*/false, a, /*neg_b=*/false, b,
            /*c_mod=*/(short)0, acc, /*reuse_a=*/false, /*reuse_b=*/false);
    }
#else
    // VALU fallback producing the same per-lane acc layout on the useful lanes:
    // acc[base+ss] = sum_d W[j][d,ss] * V[i][d,t]   (t = lane&3)
    {
        const int base = (sub & 1) ? 4 : 0;
        for (int d = 0; d < D_DIM; ++d) {
            const float vv = Vrow[d * S_DIM + s];
            #pragma unroll
            for (int ss = 0; ss < 4; ++ss)
                acc[base + ss] += Wrow[d * S_DIM + ss] * vv;
        }
    }
#endif

    // ---- epilogue: bias + logsumexp over the 4x4 block of this pair ----
    // Useful lanes hold column t = lane&3 of pair `sub` in acc[base..base+3]
    // (C/D layout: VGPR r, lanes 0-15 -> M=r; lanes 16-31 -> M=r+8).
    const bool hi = (sub & 1);
    const float c0 = hi ? acc[4] : acc[0];
    const float c1 = hi ? acc[5] : acc[1];
    const float c2 = hi ? acc[6] : acc[2];
    const float c3 = hi ? acc[7] : acc[3];

    const v4f  wb4 = *(const v4f*)(wb + (size_t)jIdx * S_DIM);  // wb[j][0..3]
    const float vbt = vb[(size_t)iIdx * S_DIM + s];             // vb[i][t]

    const float e0 = c0 + wb4.x + vbt;
    const float e1 = c1 + wb4.y + vbt;
    const float e2 = c2 + wb4.z + vbt;
    const float e3 = c3 + wb4.w + vbt;

    // 16-way logsumexp: 4 in-lane + 4-lane cross reduction (aligned groups of 4)
    float m = fmaxf(fmaxf(e0, e1), fmaxf(e2, e3));
    m = fmaxf(m, __shfl_xor(m, 1));
    m = fmaxf(m, __shfl_xor(m, 2));

    float sum = __expf(e0 - m) + __expf(e1 - m) + __expf(e2 - m) + __expf(e3 - m);
    sum += __shfl_xor(sum, 1);
    sum += __shfl_xor(sum, 2);

    // Diagonal blocks live at: khalf==0 for pairs 0,1 (lanes 0-7),
    // khalf==1 for pairs 2,3 (lanes 24-31). Column-0 lane of each writes.
    const bool useful = (khalf == ((lane >> 3) & 1));
    if (useful && s == 0 && (g + sub) < B)
        out[g + sub] = __logf(sum) + m;
}

extern "C" void kernel_launch(void* const* d_in, const int* in_sizes, int n_in,
                              void* d_out, int out_size, void* d_ws, size_t ws_size,
                              hipStream_t stream)
{
    const float* V  = (const float*)d_in[0];   // (50000, 300, 4) f32
    const float* W  = (const float*)d_in[1];   // (50000, 300, 4) f32
    const float* vb = (const float*)d_in[2];   // (50000, 4) f32
    const float* wb = (const float*)d_in[3];   // (50000, 4) f32
    const int*   IJ = (const int*)d_in[4];     // (B, 2) int32
    float* out = (float*)d_out;                // (B,) f32

    const int B = in_sizes[4] / 2;
    const int pairsPerBlock = 32;              // 8 waves x 4 pairs
    const int grid = (B + pairsPerBlock - 1) / pairsPerBlock;

    multisense_wmma_kernel<<<grid, 256, 0, stream>>>(V, W, vb, wb, IJ, out, B);
}